// FastWeightDecoder_70531952935636
// MI455X (gfx1250) — compile-verified
//
#include <hip/hip_runtime.h>
#include <hip/hip_bf16.h>

// Problem constants (from reference)
#define NN      20000
#define KNBR    32
#define INPUT   256
#define LATENT  128

typedef __attribute__((ext_vector_type(2))) float v2f;
typedef __attribute__((ext_vector_type(4))) float v4f;
typedef __attribute__((ext_vector_type(8))) float v8f;

// ---------------------------------------------------------------------------
// Kernel 1: C[M, Ncols] = A[M, Kdim] @ B[Ncols, Kdim]^T (+ bias[Ncols])
// One wave computes one 16x16 tile of C with V_WMMA_F32_16X16X4_F32.
//
// fp32 WMMA operand layout (ISA 7.12.2, 32-bit A 16x4 / B 4x16 / C 16x16):
//   A: lane 0-15 -> row M=lane, VGPR0=K0, VGPR1=K1 ; lane 16-31 -> K2,K3
//   B: lane 0-15 -> col N=lane, VGPR0=K0, VGPR1=K1 ; lane 16-31 -> K2,K3
//   C/D: col N = lane&15 ; VGPR j -> row M = j + 8*(lane>>4)
// All dims are multiples of 16 -> EXEC stays all-ones (WMMA requirement).
// ---------------------------------------------------------------------------
__global__ void __launch_bounds__(256)
fwd_gemm_nt_wmma(const float* __restrict__ A, const float* __restrict__ B,
                 const float* __restrict__ bias, float* __restrict__ C,
                 int Kdim, int Ncols) {
    const int lane  = threadIdx.x & 31;
    const int wave  = threadIdx.x >> 5;
    const int tile  = blockIdx.x * (blockDim.x >> 5) + wave;

    const int colTiles = Ncols >> 4;
    const int rowT = tile / colTiles;
    const int colT = tile - rowT * colTiles;
    const int m0 = rowT << 4;
    const int n0 = colT << 4;

    const int half = lane >> 4;   // 0: K0/K1 side, 1: K2/K3 side
    const int l15  = lane & 15;   // A row / B col within the tile

    const float* __restrict__ arow = A + (size_t)(m0 + l15) * Kdim;
    const float* __restrict__ brow = B + (size_t)(n0 + l15) * Kdim;

    v8f acc = {};
    for (int k0 = 0; k0 < Kdim; k0 += 4) {
        const int kk = k0 + 2 * half;
        v2f a; a.x = arow[kk]; a.y = arow[kk + 1];
        v2f b; b.x = brow[kk]; b.y = brow[kk + 1];
        // 8 args: (neg_a, A, neg_b, B, c_mod, C, reuse_a, reuse_b)
        acc = __builtin_amdgcn_wmma_f32_16x16x4_f32(
            false, a, false, b, (short)0, acc, false, false);
    }

    const float bcol = bias ? bias[n0 + l15] : 0.0f;

    float* __restrict__ crow = C + (size_t)(m0 + 8 * half) * Ncols + n0 + l15;
#pragma unroll
    for (int j = 0; j < 8; ++j) {
        crow[(size_t)j * Ncols] = acc[j] + bcol;
    }
}

// ---------------------------------------------------------------------------
// Kernel 2: fused gather + row-sum + rank-1 scale. One block per node n.
//   out_keys[n,k,:]   = keys[nbr[n,k], :]                (128 f32, L2 gather)
//   s                 = sum(out_keys[n,k,:])             (wave32 reduction)
//   out_values[n,k,:] = s * v[n,:] + bm                  (256 f32)
// 8 waves/block, each wave owns 4 neighbors. ~1 GB of output -> non-temporal
// stores so the streamed result doesn't thrash L2 (keys table stays resident).
// ---------------------------------------------------------------------------
__global__ void __launch_bounds__(256)
fwd_stream(const float* __restrict__ keys, const float* __restrict__ v,
           const float* __restrict__ bm, const int* __restrict__ nbr,
           float* __restrict__ out_keys, float* __restrict__ out_vals) {
    const int n    = blockIdx.x;
    const int lane = threadIdx.x & 31;
    const int wave = threadIdx.x >> 5;

    // Per-lane slice of v[n] and bm (8 floats each), reused for 4 neighbors.
    const v4f* __restrict__ v4  = (const v4f*)(v + (size_t)n * INPUT);
    const v4f* __restrict__ bm4 = (const v4f*)bm;
    const v4f va = v4[lane * 2],  vb = v4[lane * 2 + 1];
    const v4f ba = bm4[lane * 2], bb = bm4[lane * 2 + 1];

#pragma unroll
    for (int j = 0; j < 4; ++j) {
        const int k  = wave * 4 + j;
        const int nb = nbr[(size_t)n * KNBR + k];

        // Gather one key row (512 B): L2-resident table.
        const v4f kv = ((const v4f*)(keys + (size_t)nb * LATENT))[lane];

        // Stream it out (non-temporal: pure streaming output).
        v4f* __restrict__ ok =
            (v4f*)(out_keys + ((size_t)n * KNBR + k) * LATENT);
        __builtin_nontemporal_store(kv, ok + lane);

        // Wave32 row-sum.
        float s = kv.x + kv.y + kv.z + kv.w;
#pragma unroll
        for (int off = 16; off > 0; off >>= 1)
            s += __shfl_xor(s, off, 32);

        // out_values row = s * v[n] + bm (rank-1 structure of fast weights).
        v4f oa, ob;
        oa.x = fmaf(s, va.x, ba.x); oa.y = fmaf(s, va.y, ba.y);
        oa.z = fmaf(s, va.z, ba.z); oa.w = fmaf(s, va.w, ba.w);
        ob.x = fmaf(s, vb.x, bb.x); ob.y = fmaf(s, vb.y, bb.y);
        ob.z = fmaf(s, vb.z, bb.z); ob.w = fmaf(s, vb.w, bb.w);

        v4f* __restrict__ ov =
            (v4f*)(out_vals + ((size_t)n * KNBR + k) * INPUT);
        __builtin_nontemporal_store(oa, ov + lane * 2);
        __builtin_nontemporal_store(ob, ov + lane * 2 + 1);
    }
}

extern "C" void kernel_launch(void* const* d_in, const int* in_sizes, int n_in,
                              void* d_out, int out_size, void* d_ws, size_t ws_size,
                              hipStream_t stream) {
    // setup_inputs order: z, x, Wk, bk, Wm, bm, neighbors
    const float* z   = (const float*)d_in[0];  // [N, 128]
    const float* x   = (const float*)d_in[1];  // [N, 256]
    const float* Wk  = (const float*)d_in[2];  // [128, 256]
    const float* bk  = (const float*)d_in[3];  // [128]
    const float* Wm  = (const float*)d_in[4];  // [256, 128]
    const float* bm  = (const float*)d_in[5];  // [256]
    const int*   nbr = (const int*)d_in[6];    // [N, 32]

    float* out_keys = (float*)d_out;                               // [N,32,128]
    float* out_vals = out_keys + (size_t)NN * KNBR * LATENT;       // [N,32,256]

    // Workspace: keys [N,128] (10.24 MB) then v [N,256] (20.48 MB)
    float* keys = (float*)d_ws;
    float* vtab = keys + (size_t)NN * LATENT;

    // keys = x @ Wk^T + bk : tiles = (20000/16)*(128/16) = 10000 ; 8 waves/blk
    fwd_gemm_nt_wmma<<<10000 / 8, 256, 0, stream>>>(x, Wk, bk, keys,
                                                    INPUT, LATENT);
    // v = z @ Wm^T (no bias; bm folded into streaming kernel)
    // tiles = (20000/16)*(256/16) = 20000 ; 8 waves/blk
    fwd_gemm_nt_wmma<<<20000 / 8, 256, 0, stream>>>(z, Wm, nullptr, vtab,
                                                    LATENT, INPUT);
    // Fused gather + reduce + rank-1 broadcast, one block per node.
    fwd_stream<<<NN, 256, 0, stream>>>(keys, vtab, bm, nbr, out_keys, out_vals);
}